// SSIMLoss_21071109554810
// MI455X (gfx1250) — compile-verified
//
#include <hip/hip_runtime.h>
#include <hip/hip_bf16.h>

typedef float v2f __attribute__((ext_vector_type(2)));
typedef float v8f __attribute__((ext_vector_type(8)));

#define WSZ   11
#define PAD   5
#define TILE  16
#define IN_T  26            // TILE + WSZ - 1
#define HROWS 28            // IN_T padded to K-multiple of 4 (rows 26,27 zeroed)
#define PITCH 28            // padded LDS pitch for x/y tiles (conflict-free)
#define NW    4             // waves per block
#define IMG_H 512
#define IMG_W 512
#define NPLANE 96           // 32 * 3
#define TGRID 32            // 512 / 16
#define NTILES (NPLANE * TGRID * TGRID)
#define NBLOCKS ((NTILES + NW - 1) / NW)
#define NPIX (32.0f * 3.0f * 512.0f * 512.0f)

// 1-D Gaussian, sigma=1.5, window 11, normalized (matches reference)
__constant__ float GW[WSZ] = {
    0.00102838f, 0.00759877f, 0.03600077f, 0.10936069f, 0.21300539f,
    0.26601172f,
    0.21300539f, 0.10936069f, 0.03600077f, 0.00759877f, 0.00102838f
};

#define HSTRIDE (HROWS * TILE)                          // 448 floats per signal
// Per-wave LDS floats: sx(26*28) + sy(26*28) + h(5*28*16)
#define WAVE_LDS (2 * IN_T * PITCH + 5 * HSTRIDE)       // 3696 floats

__global__ __launch_bounds__(128) void ssim_main_kernel(
    const float* __restrict__ X, const float* __restrict__ Y,
    float* __restrict__ partial)
{
    __shared__ float lds[NW * WAVE_LDS];
    __shared__ float wsum[NW];

    const int lane = threadIdx.x & 31;
    const int wid  = threadIdx.x >> 5;

    float* wl = lds + wid * WAVE_LDS;
    float* sx = wl;                         // [26][28]
    float* sy = wl + IN_T * PITCH;          // [26][28]
    float* hh = wl + 2 * IN_T * PITCH;      // [5][28][16] (rows 26,27 zero)

    const int tidx   = blockIdx.x * NW + wid;
    const bool active = (tidx < NTILES);
    const int t  = active ? tidx : 0;
    const int plane = t >> 10;              // / (32*32)
    const int tt = t & 1023;
    const int ty = tt >> 5, tx = tt & 31;

    const float* xp = X + (size_t)plane * (IMG_H * IMG_W);
    const float* yp = Y + (size_t)plane * (IMG_H * IMG_W);
    const int r0 = ty * TILE - PAD;
    const int c0 = tx * TILE - PAD;

    // ---- Stage 26x26 windows of x and y into LDS (zero-padded at edges) ----
    for (int i = lane; i < IN_T * IN_T; i += 32) {
        int r = i / IN_T, c = i - r * IN_T;
        int gr = r0 + r, gc = c0 + c;
        bool ok = (gr >= 0) & (gr < IMG_H) & (gc >= 0) & (gc < IMG_W);
        size_t off = (size_t)gr * IMG_W + gc;
        sx[r * PITCH + c] = ok ? xp[off] : 0.0f;
        sy[r * PITCH + c] = ok ? yp[off] : 0.0f;
    }
    __syncthreads();

    // ---- Horizontal 11-tap pass for 5 signals: x, y, x^2, y^2, xy ----
    // Rows 26,27 of each signal are written as zero so the vertical-pass
    // B-chunk loads (K padded to 28) need no conditionals.
    for (int i = lane; i < HROWS * TILE; i += 32) {
        int r = i >> 4, c = i & 15;
        float hx = 0.f, hy = 0.f, hxx = 0.f, hyy = 0.f, hxy = 0.f;
        if (r < IN_T) {
            const float* rx = sx + r * PITCH + c;
            const float* ry = sy + r * PITCH + c;
            #pragma unroll
            for (int k = 0; k < WSZ; ++k) {
                float wk = GW[k];
                float ax = rx[k], ay = ry[k];
                hx  += wk * ax;       hy  += wk * ay;
                hxx += wk * ax * ax;  hyy += wk * ay * ay;
                hxy += wk * ax * ay;
            }
        }
        hh[0 * HSTRIDE + i] = hx;
        hh[1 * HSTRIDE + i] = hy;
        hh[2 * HSTRIDE + i] = hxx;
        hh[3 * HSTRIDE + i] = hyy;
        hh[4 * HSTRIDE + i] = hxy;
    }
    __syncthreads();

    // ---- Vertical pass as D = Band(16x28) x H(28x16) via f32 WMMA ----
    const int half = lane >> 4;     // lane half selects K sub-pair
    const int mrow = lane & 15;     // A: M row ; B/D: N column

    // A band chunks: A[M][j] = GW[j-M] if 0<=j-M<11 and j<26 else 0
    v2f A[7];
    #pragma unroll
    for (int q = 0; q < 7; ++q) {
        int j0 = 4 * q + 2 * half;
        int d0 = j0 - mrow, d1 = j0 + 1 - mrow;
        float a0 = (j0 < IN_T     && d0 >= 0 && d0 < WSZ) ? GW[d0] : 0.0f;
        float a1 = (j0 + 1 < IN_T && d1 >= 0 && d1 < WSZ) ? GW[d1] : 0.0f;
        A[q][0] = a0; A[q][1] = a1;
    }

    v8f acc[5];
    #pragma unroll
    for (int s = 0; s < 5; ++s) {
        const float* hs = hh + s * HSTRIDE;
        v8f c8 = {0.f, 0.f, 0.f, 0.f, 0.f, 0.f, 0.f, 0.f};
        #pragma unroll
        for (int q = 0; q < 7; ++q) {
            int j0 = 4 * q + 2 * half;          // 0..26, always < HROWS
            v2f B;
            B[0] = hs[j0 * TILE + mrow];
            B[1] = hs[(j0 + 1) * TILE + mrow];
            // D = A*B + C : v_wmma_f32_16x16x4_f32
            c8 = __builtin_amdgcn_wmma_f32_16x16x4_f32(
                false, A[q], false, B, (short)0, c8, false, false);
        }
        acc[s] = c8;
    }

    // ---- SSIM map (elementwise on D layout) + per-lane partial sum ----
    const float C1 = 1.0e-4f;   // (0.01*1.0)^2
    const float C2 = 9.0e-4f;   // (0.03*1.0)^2
    float psum = 0.0f;
    #pragma unroll
    for (int v = 0; v < 8; ++v) {
        float mx  = acc[0][v], my  = acc[1][v];
        float ex2 = acc[2][v], ey2 = acc[3][v], exy = acc[4][v];
        float mx2 = mx * mx, my2 = my * my, mxy = mx * my;
        float vx  = ex2 - mx2, vy = ey2 - my2, vxy = exy - mxy;
        float num = (2.0f * mxy + C1) * (2.0f * vxy + C2);
        float den = (mx2 + my2 + C1) * (vx + vy + C2);
        psum += num / den;
    }
    if (!active) psum = 0.0f;

    // ---- Deterministic reduction: wave shuffle tree -> block -> partial ----
    #pragma unroll
    for (int off = 16; off > 0; off >>= 1)
        psum += __shfl_xor(psum, off, 32);
    if (lane == 0) wsum[wid] = psum;
    __syncthreads();
    if (threadIdx.x == 0) {
        float b = 0.0f;
        #pragma unroll
        for (int i = 0; i < NW; ++i) b += wsum[i];
        partial[blockIdx.x] = b;
    }
}

__global__ __launch_bounds__(256) void ssim_reduce_kernel(
    const float* __restrict__ partial, int n, float* __restrict__ out)
{
    __shared__ float sh[256];
    float s = 0.0f;
    for (int i = threadIdx.x; i < n; i += 256) s += partial[i];  // fixed order
    sh[threadIdx.x] = s;
    __syncthreads();
    for (int st = 128; st > 0; st >>= 1) {
        if (threadIdx.x < st) sh[threadIdx.x] += sh[threadIdx.x + st];
        __syncthreads();
    }
    if (threadIdx.x == 0) out[0] = 1.0f - sh[0] / NPIX;
}

extern "C" void kernel_launch(void* const* d_in, const int* in_sizes, int n_in,
                              void* d_out, int out_size, void* d_ws, size_t ws_size,
                              hipStream_t stream) {
    (void)in_sizes; (void)n_in; (void)out_size; (void)ws_size;
    const float* x = (const float*)d_in[0];
    const float* y = (const float*)d_in[1];
    float* out     = (float*)d_out;
    float* partial = (float*)d_ws;          // NBLOCKS floats (96 KB) of scratch

    ssim_main_kernel<<<NBLOCKS, 128, 0, stream>>>(x, y, partial);
    ssim_reduce_kernel<<<1, 256, 0, stream>>>(partial, NBLOCKS, out);
}